// GeometryPostProcessor_25074019074088
// MI455X (gfx1250) — compile-verified
//
#include <hip/hip_runtime.h>
#include <hip/hip_bf16.h>
#include <stdint.h>
#include <math.h>

// Problem constants (match reference setup_inputs)
#define Bn 32
#define Cn 3
#define Hn 256
#define Wn 832
#define HWn (Hn * Wn)
#define Kn 100            // MAX_DET
#define CHUNK 16          // rows per TDM tile (plus halo)
#define NCHUNK (Hn / CHUNK)
#define NBINS 1024
#define CAND 512
#define TILE_ROWS (CHUNK + 2)
#define TILE_BYTES (TILE_ROWS * Wn * 4)

#define PI_F 3.14159265358979323846f
#define DEPTH_MEAN_F 28.01f
#define DET_THRESHOLD_F 0.25f

typedef unsigned int u32x4 __attribute__((ext_vector_type(4)));
typedef int i32x4 __attribute__((ext_vector_type(4)));
typedef int i32x8 __attribute__((ext_vector_type(8)));

#if __has_builtin(__builtin_amdgcn_tensor_load_to_lds) && __has_builtin(__builtin_amdgcn_s_wait_tensorcnt)
#define HAS_TDM 1
#else
#define HAS_TDM 0
#endif

#if HAS_TDM
// Issue one TDM DMA: a 2D tile of `nrows` rows x Wn fp32 from global into LDS
// at byte offset lds_off. D# layout per CDNA5 ISA.
// 6-arg builtin on this toolchain:
//   (u32x4 group0, i32x8 group1, i32x4 group2, i32x4 group3, i32x8, i32 cpol)
// Groups 2/3 (and trailing i32x8) only matter for >2D tensors; zeros for 2D.
__device__ __forceinline__ void tdm_load_rows(const void* gptr, int nrows, unsigned lds_off) {
  unsigned long long ga = (unsigned long long)(uintptr_t)gptr;
  u32x4 g0;
  g0[0] = 1u;                                            // count=1, user descriptor
  g0[1] = lds_off;                                       // lds_addr (bytes)
  g0[2] = (unsigned)(ga & 0xFFFFFFFFull);                // global_addr[31:0]
  g0[3] = (unsigned)((ga >> 32) & 0x01FFFFFFull) | 0x80000000u; // addr[56:32] | type=2
  i32x8 g1;
  g1[0] = 0x00020000;                                    // data_size=2 (4 bytes), mask=0
  g1[1] = (Wn & 0xFFFF) << 16;                           // tensor_dim0 lo16 (bits 63:48)
  g1[2] = ((Wn >> 16) & 0xFFFF) | ((nrows & 0xFFFF) << 16); // dim0 hi16 | tensor_dim1 lo16
  g1[3] = ((nrows >> 16) & 0xFFFF) | ((Wn & 0xFFFF) << 16); // dim1 hi16 | tile_dim0
  g1[4] = (nrows & 0xFFFF);                              // tile_dim1 (tile_dim2=0 -> 2D)
  g1[5] = Wn;                                            // tensor_dim0_stride lo32
  g1[6] = 0;
  g1[7] = 0;
  i32x4 z4 = (i32x4){0, 0, 0, 0};
  i32x8 z8 = (i32x8){0, 0, 0, 0, 0, 0, 0, 0};
  __builtin_amdgcn_tensor_load_to_lds(g0, g1, z4, z4, z8, 0);
}
#endif

// Descending bitonic sort of 512 (value, payload) pairs in LDS, 256 threads.
__device__ __forceinline__ void bitonic_desc_512(float* v, int* idx, int tid) {
  for (unsigned ksz = 2; ksz <= CAND; ksz <<= 1) {
    for (unsigned j = ksz >> 1; j > 0; j >>= 1) {
      for (unsigned t = (unsigned)tid; t < CAND; t += 256) {
        unsigned x = t ^ j;
        if (x > t) {
          bool desc_seg = ((t & ksz) == 0);
          float a = v[t], b = v[x];
          bool sw = desc_seg ? (a < b) : (a > b);
          if (sw) {
            v[t] = b; v[x] = a;
            int q = idx[t]; idx[t] = idx[x]; idx[x] = q;
          }
        }
      }
      __syncthreads();
    }
  }
}

__device__ __forceinline__ void chunk_geom(int ch, int& r0, int& g0r, int& nrows) {
  r0 = ch * CHUNK;
  g0r = (r0 > 0) ? r0 - 1 : 0;
  const int g1r = (r0 + CHUNK + 1 < Hn) ? r0 + CHUNK + 1 : Hn;
  nrows = g1r - g0r;
}

// Double-buffered TDM scan of one (b,c) plane with on-the-fly 3x3 NMS.
// MODE 0: accumulate histogram of kept scores.
// MODE 1: collect (score, index) candidates with bin >= thr into cval/cidx.
template <int MODE>
__device__ void scan_plane(const float* __restrict__ plane,
                           float* __restrict__ tile0, float* __restrict__ tile1,
                           unsigned* __restrict__ hist,
                           float* __restrict__ cval, int* __restrict__ cidx,
                           int* __restrict__ misc, int thr, int tid) {
  int r0, g0r, nrows;
#if HAS_TDM
  // Prologue: kick off chunk 0 into buffer 0.
  chunk_geom(0, r0, g0r, nrows);
  if (tid == 0) tdm_load_rows(plane + (size_t)g0r * Wn, nrows, 0u);
#endif
  for (int ch = 0; ch < NCHUNK; ++ch) {
    chunk_geom(ch, r0, g0r, nrows);
#if HAS_TDM
    const bool has_next = (ch + 1 < NCHUNK);
    if (has_next) {
      int nr0, ng0r, nnrows;
      chunk_geom(ch + 1, nr0, ng0r, nnrows);
      const unsigned lds_off = ((ch + 1) & 1) ? (unsigned)TILE_BYTES : 0u;
      if (tid == 0) tdm_load_rows(plane + (size_t)ng0r * Wn, nnrows, lds_off);
      // Two DMAs in flight on wave 0; wait until the oldest (this chunk) lands.
      __builtin_amdgcn_s_wait_tensorcnt(1);
    } else {
      __builtin_amdgcn_s_wait_tensorcnt(0);
    }
    float* tile = (ch & 1) ? tile1 : tile0;
#else
    float* tile = (ch & 1) ? tile1 : tile0;
    for (int i = tid; i < nrows * Wn; i += 256) tile[i] = plane[(size_t)g0r * Wn + i];
#endif
    __syncthreads();  // tile ready for all waves
    for (int i = tid; i < CHUNK * Wn; i += 256) {
      const int rr = r0 + i / Wn;
      const int cc = i % Wn;
      const float v = tile[(rr - g0r) * Wn + cc];
      float m = v;
#pragma unroll
      for (int dr = -1; dr <= 1; ++dr) {
        const int gr = rr + dr;
        if (gr < 0 || gr >= Hn) continue;
        const int lrr = gr - g0r;
#pragma unroll
        for (int dc = -1; dc <= 1; ++dc) {
          const int gc = cc + dc;
          if (gc < 0 || gc >= Wn) continue;
          m = fmaxf(m, tile[lrr * Wn + gc]);
        }
      }
      if (v == m) {  // survives NMS (ties kept, as in reference)
        int bin = (int)(v * (float)NBINS);
        bin = bin < 0 ? 0 : (bin > NBINS - 1 ? NBINS - 1 : bin);
        if (MODE == 0) {
          atomicAdd(&hist[bin], 1u);
        } else if (bin >= thr) {
          const int p = atomicAdd(&misc[0], 1);
          if (p < CAND) { cval[p] = v; cidx[p] = rr * Wn + cc; }
        }
      }
    }
    __syncthreads();  // compute on this buffer done before it is re-DMA'd
  }
}

// ---------------------------------------------------------------------------
// Kernel 1: fused 3x3 maxpool-NMS + exact per-(b,c) top-K via histogram select.
// One block per (b,c) plane; TDM double-buffers heatmap tiles into LDS.
// ---------------------------------------------------------------------------
__global__ __launch_bounds__(256) void nms_topk_kernel(const float* __restrict__ heat,
                                                       float* __restrict__ scores_c,
                                                       int* __restrict__ inds_c) {
  extern __shared__ unsigned char smem[];
  float* tile0   = (float*)smem;                                       // TILE_ROWS x Wn
  float* tile1   = (float*)(smem + TILE_BYTES);                        // TILE_ROWS x Wn
  unsigned* hist = (unsigned*)(smem + 2 * TILE_BYTES);                 // NBINS
  float* cval    = (float*)(smem + 2 * TILE_BYTES + NBINS * 4);        // CAND
  int* cidx      = (int*)(smem + 2 * TILE_BYTES + NBINS * 4 + CAND * 4);
  int* misc      = (int*)(smem + 2 * TILE_BYTES + NBINS * 4 + CAND * 8); // [0]=cnt [1]=thr

  const int tid = threadIdx.x;
  const int bc = blockIdx.x;
  const float* plane = heat + (size_t)bc * HWn;

  for (int i = tid; i < NBINS; i += 256) hist[i] = 0u;
  if (tid == 0) misc[0] = 0;
  __syncthreads();

  // pass 0: histogram of kept (local-max) scores
  scan_plane<0>(plane, tile0, tile1, hist, cval, cidx, misc, 0, tid);
  __syncthreads();

  // threshold bin: cumulative-from-top crosses K
  if (tid == 0) {
    unsigned cum = 0;
    int tb = 0;
    for (int i = NBINS - 1; i >= 0; --i) {
      cum += hist[i];
      if (cum >= (unsigned)Kn) { tb = i; break; }
    }
    misc[1] = tb;
    misc[0] = 0;
  }
  __syncthreads();
  const int thr = misc[1];

  // pass 1: collect candidates >= threshold bin (plane is L2-resident now)
  scan_plane<1>(plane, tile0, tile1, hist, cval, cidx, misc, thr, tid);
  __syncthreads();

  int nc = misc[0];
  if (nc > CAND) nc = CAND;
  for (int i = tid; i < CAND; i += 256) {
    if (i >= nc) { cval[i] = -1e30f; cidx[i] = 0; }
  }
  __syncthreads();
  bitonic_desc_512(cval, cidx, tid);
  for (int k = tid; k < Kn; k += 256) {
    const bool ok = (k < nc);
    scores_c[bc * Kn + k] = ok ? cval[k] : 0.0f;
    inds_c[bc * Kn + k] = ok ? cidx[k] : 0;
  }
}

// ---------------------------------------------------------------------------
// Kernel 2: per-batch top-K over the C*K=300 class candidates.
// ---------------------------------------------------------------------------
__global__ __launch_bounds__(256) void merge_topk_kernel(const float* __restrict__ scores_c,
                                                         const int* __restrict__ inds_c,
                                                         float* __restrict__ det_score,
                                                         int* __restrict__ det_cls,
                                                         int* __restrict__ det_ind) {
  __shared__ float sv[CAND];
  __shared__ int ss[CAND];
  const int tid = threadIdx.x;
  const int b = blockIdx.x;
  for (int i = tid; i < CAND; i += 256) {
    if (i < Cn * Kn) { sv[i] = scores_c[b * Cn * Kn + i]; ss[i] = i; }
    else             { sv[i] = -1e30f; ss[i] = -1; }
  }
  __syncthreads();
  bitonic_desc_512(sv, ss, tid);
  for (int k = tid; k < Kn; k += 256) {
    const int s = ss[k];
    det_score[b * Kn + k] = (s >= 0) ? sv[k] : 0.0f;
    det_cls[b * Kn + k] = (s >= 0) ? (s / Kn) : 0;
    det_ind[b * Kn + k] = (s >= 0) ? inds_c[b * Cn * Kn + s] : 0;
  }
}

// ---------------------------------------------------------------------------
// Kernel 3: per-detection geometry (3200 threads).
// ---------------------------------------------------------------------------
__global__ __launch_bounds__(256) void geometry_kernel(const float* __restrict__ reg,
                                                       const float* __restrict__ trans,
                                                       const float* __restrict__ Kmat,
                                                       const float* __restrict__ sizev,
                                                       const float* __restrict__ hcam,
                                                       const float* __restrict__ dimsIn,
                                                       const float* __restrict__ det_score,
                                                       const int* __restrict__ det_cls,
                                                       const int* __restrict__ det_ind,
                                                       float* __restrict__ out) {
  const int i = blockIdx.x * 256 + threadIdx.x;
  if (i >= Bn * Kn) return;
  const int b = i / Kn;
  const float score = det_score[i];
  const float clsf = (float)det_cls[i];
  const int ind = det_ind[i];
  const float xs = (float)(ind % Wn);
  const float ys = (float)(ind / Wn);

  const float* rb = reg + (size_t)b * 4 * HWn;
  const float log_dv_delta = rb[ind];
  const float off_u = rb[HWn + ind];
  const float ori0 = rb[2 * HWn + ind];
  const float ori1 = rb[3 * HWn + ind];

  // trans_mat^{-1} via adjugate (per batch, tiny)
  const float* T = trans + (size_t)b * 9;
  const float a = T[0], bb = T[1], c = T[2];
  const float d = T[3], e = T[4], f = T[5];
  const float g = T[6], hh = T[7], ii = T[8];
  const float c00 = e * ii - f * hh, c01 = -(d * ii - f * g), c02 = d * hh - e * g;
  const float c10 = -(bb * ii - c * hh), c11 = a * ii - c * g, c12 = -(a * hh - bb * g);
  const float c20 = bb * f - c * e, c21 = -(a * f - c * d), c22 = a * e - bb * d;
  const float det = a * c00 + bb * c01 + c * c02;
  const float rdet = 1.0f / det;
  const float px0 = xs + off_u, py0 = ys;
  const float img_x = (c00 * px0 + c10 * py0 + c20) * rdet;
  const float img_y = (c01 * px0 + c11 * py0 + c21) * rdet;
  (void)img_y; (void)c12; (void)c22; (void)c02;

  const float* Km = Kmat + (size_t)b * 9;
  const float fx = Km[0], k01 = Km[1], cx = Km[2];
  const float k10 = Km[3], fy = Km[4], cy = Km[5];
  const float k20 = Km[6], k21 = Km[7], k22 = Km[8];
  const float h = hcam[b];
  float d0 = dimsIn[b * 3 + 0], d1 = dimsIn[b * 3 + 1], d2 = dimsIn[b * 3 + 2];
  d0 = isfinite(d0) ? d0 : 3.88f;
  d1 = isfinite(d1) ? d1 : 1.63f;
  d2 = isfinite(d2) ? d2 : 1.53f;

  const float h_ref = h - d1 * 0.5f;
  const float fyh = fy * fabsf(h_ref);
  const float log_dv_ref = logf(fmaxf(fyh, 1e-7f) / DEPTH_MEAN_F);
  const float log_dv = fminf(fmaxf(log_dv_ref + log_dv_delta, -4.0f), 8.0f);
  const float depth = fminf(fmaxf(fyh * expf(-log_dv), 0.5f), 120.0f);
  const float locx = (img_x - cx) * depth / fx;

  const float ray = atanf(locx / (depth + 1e-7f));
  float alpha = atanf(ori0 / (ori1 + 1e-7f));
  alpha += (ori1 >= 0.0f) ? (-PI_F * 0.5f) : (PI_F * 0.5f);
  float roty = alpha + ray;
  if (roty > PI_F) roty -= 2.0f * PI_F;
  if (roty < -PI_F) roty += 2.0f * PI_F;
  const float cr = cosf(roty), sr = sinf(roty);

  const float cx8[8] = {-0.5f, 0.5f, 0.5f, 0.5f, 0.5f, -0.5f, -0.5f, -0.5f};
  const float cy8[8] = {-1.0f, -1.0f, 0.0f, 0.0f, -1.0f, -1.0f, 0.0f, 0.0f};
  const float cz8[8] = {-0.5f, -0.5f, -0.5f, 0.5f, 0.5f, 0.5f, 0.5f, -0.5f};
  float umin = 1e30f, umax = -1e30f, vmin = 1e30f, vmax = -1e30f;
#pragma unroll
  for (int j = 0; j < 8; ++j) {
    const float xc = d0 * cx8[j], yc = d1 * cy8[j], zc = d2 * cz8[j];
    const float X = cr * xc + sr * zc + locx;
    const float Y = yc + h;
    const float Z = -sr * xc + cr * zc + depth;
    const float pu = fx * X + k01 * Y + cx * Z;
    const float pv = k10 * X + fy * Y + cy * Z;
    const float pw = k20 * X + k21 * Y + k22 * Z;
    const float uu = pu / pw, vv = pv / pw;
    umin = fminf(umin, uu); umax = fmaxf(umax, uu);
    vmin = fminf(vmin, vv); vmax = fmaxf(vmax, vv);
  }
  const float img_w = sizev[0], img_h = sizev[1];
  const float xminc = fminf(fmaxf(umin, 0.0f), img_w);
  const float xmaxc = fminf(fmaxf(umax, 0.0f), img_w);
  const float yminc = fminf(fmaxf(vmin, 0.0f), img_h);
  const float ymaxc = fminf(fmaxf(vmax, 0.0f), img_h);

  const float keep = (score > DET_THRESHOLD_F) ? 1.0f : 0.0f;
  float* o = out + (size_t)i * 14;
  o[0]  = clsf * keep;
  o[1]  = alpha * keep;
  o[2]  = xminc * keep;
  o[3]  = yminc * keep;
  o[4]  = xmaxc * keep;
  o[5]  = ymaxc * keep;
  o[6]  = d1 * keep;   // roll(dims,-1)
  o[7]  = d2 * keep;
  o[8]  = d0 * keep;
  o[9]  = locx * keep;
  o[10] = h * keep;
  o[11] = depth * keep;
  o[12] = roty * keep;
  o[13] = score * keep;
}

extern "C" void kernel_launch(void* const* d_in, const int* in_sizes, int n_in,
                              void* d_out, int out_size, void* d_ws, size_t ws_size,
                              hipStream_t stream) {
  (void)in_sizes; (void)n_in; (void)out_size; (void)ws_size;
  const float* heat  = (const float*)d_in[0];
  const float* reg   = (const float*)d_in[1];
  const float* trans = (const float*)d_in[2];
  const float* Kmat  = (const float*)d_in[3];
  const float* sizev = (const float*)d_in[4];
  const float* hcam  = (const float*)d_in[5];
  const float* dims  = (const float*)d_in[6];

  unsigned char* ws = (unsigned char*)d_ws;
  float* scores_c  = (float*)ws;                               // B*C*K
  int*   inds_c    = (int*)(ws + (size_t)Bn * Cn * Kn * 4);    // B*C*K
  float* det_score = (float*)(ws + (size_t)Bn * Cn * Kn * 8);  // B*K
  int*   det_cls   = (int*)(ws + (size_t)Bn * Cn * Kn * 8 + (size_t)Bn * Kn * 4);
  int*   det_ind   = (int*)(ws + (size_t)Bn * Cn * Kn * 8 + (size_t)Bn * Kn * 8);

  const size_t smem1 = (size_t)2 * TILE_BYTES + NBINS * 4 + CAND * 8 + 32;
  nms_topk_kernel<<<Bn * Cn, 256, smem1, stream>>>(heat, scores_c, inds_c);
  merge_topk_kernel<<<Bn, 256, 0, stream>>>(scores_c, inds_c, det_score, det_cls, det_ind);
  geometry_kernel<<<(Bn * Kn + 255) / 256, 256, 0, stream>>>(
      reg, trans, Kmat, sizev, hcam, dims, det_score, det_cls, det_ind, (float*)d_out);
}